// NPPLoss_16192026706270
// MI455X (gfx1250) — compile-verified
//
#include <hip/hip_runtime.h>
#include <hip/hip_bf16.h>

typedef __attribute__((ext_vector_type(2))) float v2f;
typedef __attribute__((ext_vector_type(8))) float v8f;

#define NN 1024
#define NB 16
#define NBLK (NN / NB)   // 64
#define BATCH 64
#define NOISE_EPS 1e-4f

// ---------------------------------------------------------------------------
// Kernel 1: build K[b][i][j] = exp(-d2(i,j) / (2*ls^2)) + NOISE*(i==j)
// grid (64,64,64) tiles of 16x16, block (16,16)
// ---------------------------------------------------------------------------
__global__ __launch_bounds__(256) void nppk_build_K(
    const long long* __restrict__ pins, const float* __restrict__ lls,
    float* __restrict__ Kws) {
  const int b  = blockIdx.z;
  const int it = blockIdx.y * NB;   // row tile base
  const int jt = blockIdx.x * NB;   // col tile base
  __shared__ float xi[NB], yi_[NB], xj[NB], yj[NB];
  const int t = threadIdx.y * NB + threadIdx.x;
  if (t < NB) {
    const long long* p = pins + ((size_t)b * NN + it + t) * 2;
    xi[t]  = (float)p[0];
    yi_[t] = (float)p[1];
  } else if (t < 2 * NB) {
    const int u = t - NB;
    const long long* p = pins + ((size_t)b * NN + jt + u) * 2;
    xj[u] = (float)p[0];
    yj[u] = (float)p[1];
  }
  __syncthreads();
  const float scale = -0.5f * __expf(-2.0f * lls[0]);  // -1/(2*ls^2)
  const int r = threadIdx.y, c = threadIdx.x;
  const float dx = xi[r] - xj[c];
  const float dy = yi_[r] - yj[c];
  float v = __expf(scale * (dx * dx + dy * dy));
  const int gi = it + r, gj = jt + c;
  if (gi == gj) v += NOISE_EPS;
  Kws[(size_t)b * NN * NN + (size_t)gi * NN + gj] = v;
}

// ---------------------------------------------------------------------------
// Kernel 2: diff[b][i] = targets[b][i] - preds[b][pins0][pins1]
// ---------------------------------------------------------------------------
__global__ __launch_bounds__(256) void nppk_build_diff(
    const float* __restrict__ preds, const float* __restrict__ targets,
    const long long* __restrict__ pins, float* __restrict__ diffws) {
  const int b = blockIdx.y;
  const int i = blockIdx.x * blockDim.x + threadIdx.x;
  const long long* p = pins + ((size_t)b * NN + i) * 2;
  const int r = (int)p[0], c = (int)p[1];
  const float yh = preds[(size_t)b * 256 * 256 + r * 256 + c];
  diffws[(size_t)b * NN + i] = targets[(size_t)b * NN + i] - yh;
}

// ---------------------------------------------------------------------------
// Kernel 3: per-batch blocked Cholesky (NB=16) with WMMA f32 trailing update,
// forward solve L y = diff, res[b] = ||y||^2 + 2*sum(log diag(L)).
// One 8-wave workgroup per batch. Lower triangle maintained.
// Trailing update uses 2x2 register blocking: each wave owns a 32x32
// super-tile -> 16 WMMAs per 16 operand-fragment loads (2x reuse).
// ---------------------------------------------------------------------------
__global__ __launch_bounds__(256) void nppk_chol_solve(
    float* __restrict__ Kws, const float* __restrict__ diffws,
    float* __restrict__ resws) {
  const int b    = blockIdx.x;
  const int tid  = threadIdx.x;
  const int lane = tid & 31;
  const int wave = tid >> 5;
  const int nwaves = blockDim.x >> 5;   // 8
  float* A = Kws + (size_t)b * NN * NN;

  __shared__ float Ld[NB * NB];   // current diagonal block L_kk
  __shared__ float yv[NN];        // rhs / solution vector
  __shared__ float red[8];

  float logdet_acc = 0.0f;   // only meaningful on tid 0

  for (int kb = 0; kb < NBLK; ++kb) {
    const int k0 = kb * NB;

    // ---- Step 1: factor 16x16 diagonal block (serial, tiny) ----
    if (tid == 0) {
      for (int r = 0; r < NB; ++r)
        for (int c = 0; c < NB; ++c)
          Ld[r * NB + c] = (c <= r) ? A[(size_t)(k0 + r) * NN + (k0 + c)] : 0.0f;
      for (int j = 0; j < NB; ++j) {
        float s = Ld[j * NB + j];
        for (int t = 0; t < j; ++t) s -= Ld[j * NB + t] * Ld[j * NB + t];
        s = sqrtf(fmaxf(s, 1e-20f));
        Ld[j * NB + j] = s;
        logdet_acc += 2.0f * __logf(s);
        const float inv = 1.0f / s;
        for (int r = j + 1; r < NB; ++r) {
          float v = Ld[r * NB + j];
          for (int t = 0; t < j; ++t) v -= Ld[r * NB + t] * Ld[j * NB + t];
          Ld[r * NB + j] = v * inv;
        }
      }
      for (int r = 0; r < NB; ++r)
        for (int c = 0; c < NB; ++c)
          A[(size_t)(k0 + r) * NN + (k0 + c)] = Ld[r * NB + c];
    }
    __syncthreads();

    // ---- Step 2: panel TRSM: rows below diag, L[r][k0..] = A_row * Lkk^-T ----
    for (int r = k0 + NB + tid; r < NN; r += blockDim.x) {
      float x[NB];
      float* arow = &A[(size_t)r * NN + k0];
#pragma unroll
      for (int c = 0; c < NB; ++c) x[c] = arow[c];
#pragma unroll
      for (int c = 0; c < NB; ++c) {
        float v = x[c];
        for (int t = 0; t < c; ++t) v -= x[t] * Ld[c * NB + t];
        x[c] = v / Ld[c * NB + c];
      }
#pragma unroll
      for (int c = 0; c < NB; ++c) arow[c] = x[c];
    }
    __threadfence();
    __syncthreads();

    // ---- Step 3: trailing update A_ij -= P_i * P_j^T via v_wmma_f32_16x16x4
    const int T = NBLK - kb - 1;   // trailing block rows
    if (T > 0) {
      const int TS = (T + 1) >> 1;          // 32x32 super-tiles per side
      const int nsup = TS * (TS + 1) / 2;   // lower-triangle super-tiles
      const int halfsel = lane >> 4;        // 0: K-pair {0,1}, 1: {2,3}
      const int lrow    = lane & 15;
      for (int t = wave; t < nsup; t += nwaves) {
        // linear index -> (si, sj), si >= sj (lower triangle of TS x TS)
        int si = (int)((sqrtf(8.0f * (float)t + 1.0f) - 1.0f) * 0.5f);
        while ((si + 1) * (si + 2) / 2 <= t) ++si;
        while (si * (si + 1) / 2 > t) --si;
        const int sj = t - si * (si + 1) / 2;

        // block-row/col bases; out-of-range second row/col clamped to the
        // first (loads stay in-bounds, results masked at store time).
        const bool rowb = (2 * si + 1) < T;   // wave-uniform
        const bool colb = (2 * sj + 1) < T;   // wave-uniform
        const int ia = k0 + NB * (1 + 2 * si);
        const int ib = rowb ? ia + NB : ia;
        const int ja = k0 + NB * (1 + 2 * sj);
        const int jb = colb ? ja + NB : ja;

        v8f c00 = {}, c01 = {}, c10 = {}, c11 = {};
#pragma unroll
        for (int f = 0; f < 4; ++f) {
          // A frag (16x4): lane<16 holds K=4f+{0,1}; lane>=16 holds K=4f+{2,3}
          // B frag (4x16): same addressing on block-row j (B = P_j^T)
          const int col = k0 + 4 * f + 2 * halfsel;
          const v2f a0 = *(const v2f*)&A[(size_t)(ia + lrow) * NN + col];
          const v2f a1 = *(const v2f*)&A[(size_t)(ib + lrow) * NN + col];
          const v2f b0 = *(const v2f*)&A[(size_t)(ja + lrow) * NN + col];
          const v2f b1 = *(const v2f*)&A[(size_t)(jb + lrow) * NN + col];
          c00 = __builtin_amdgcn_wmma_f32_16x16x4_f32(
              false, a0, false, b0, (short)0, c00, false, false);
          c01 = __builtin_amdgcn_wmma_f32_16x16x4_f32(
              false, a0, false, b1, (short)0, c01, false, false);
          c10 = __builtin_amdgcn_wmma_f32_16x16x4_f32(
              false, a1, false, b0, (short)0, c10, false, false);
          c11 = __builtin_amdgcn_wmma_f32_16x16x4_f32(
              false, a1, false, b1, (short)0, c11, false, false);
        }
        // C/D layout: VGPR r -> row (r + 8*halfsel), col = lrow.
        // Store guards are wave-uniform -> EXEC stays all-ones.
#pragma unroll
        for (int r = 0; r < 8; ++r) {
          const size_t idx =
              (size_t)(ia + r + 8 * halfsel) * NN + (ja + lrow);
          A[idx] = A[idx] - c00[r];
        }
        if (si > sj) {   // upper sub-tile of diagonal super-tile skipped
#pragma unroll
          for (int r = 0; r < 8; ++r) {
            const size_t idx =
                (size_t)(ia + r + 8 * halfsel) * NN + (jb + lrow);
            A[idx] = A[idx] - c01[r];
          }
        }
        if (rowb) {
#pragma unroll
          for (int r = 0; r < 8; ++r) {
            const size_t idx =
                (size_t)(ib + r + 8 * halfsel) * NN + (ja + lrow);
            A[idx] = A[idx] - c10[r];
          }
          if (colb) {
#pragma unroll
            for (int r = 0; r < 8; ++r) {
              const size_t idx =
                  (size_t)(ib + r + 8 * halfsel) * NN + (jb + lrow);
              A[idx] = A[idx] - c11[r];
            }
          }
        }
      }
    }
    __threadfence();
    __syncthreads();
  }

  // ---- Forward solve L y = diff (column-oriented parallel axpy in LDS) ----
  const float* dvec = diffws + (size_t)b * NN;
  for (int i = tid; i < NN; i += blockDim.x) yv[i] = dvec[i];
  __syncthreads();
  for (int c = 0; c < NN; ++c) {
    if (tid == 0) yv[c] = yv[c] / A[(size_t)c * NN + c];
    __syncthreads();
    const float yc = yv[c];
    for (int r = c + 1 + tid; r < NN; r += blockDim.x)
      yv[r] -= A[(size_t)r * NN + c] * yc;
    __syncthreads();
  }

  // ---- quad = ||y||^2, reduce, write quad + logdet ----
  float part = 0.0f;
  for (int i = tid; i < NN; i += blockDim.x) part += yv[i] * yv[i];
#pragma unroll
  for (int o = 16; o > 0; o >>= 1) part += __shfl_down(part, o, 32);
  if (lane == 0) red[wave] = part;
  __syncthreads();
  if (tid == 0) {
    float q = 0.0f;
    for (int w = 0; w < 8; ++w) q += red[w];
    resws[b] = q + logdet_acc;
  }
}

// ---------------------------------------------------------------------------
// Kernel 4: out = mean over batches
// ---------------------------------------------------------------------------
__global__ void nppk_finalize(const float* __restrict__ resws,
                              float* __restrict__ out) {
  __shared__ float s[BATCH];
  s[threadIdx.x] = resws[threadIdx.x];
  __syncthreads();
  if (threadIdx.x == 0) {
    float a = 0.0f;
    for (int i = 0; i < BATCH; ++i) a += s[i];
    out[0] = a * (1.0f / BATCH);
  }
}

// ---------------------------------------------------------------------------
extern "C" void kernel_launch(void* const* d_in, const int* in_sizes, int n_in,
                              void* d_out, int out_size, void* d_ws,
                              size_t ws_size, hipStream_t stream) {
  const float* preds       = (const float*)d_in[0];      // (64,256,256) f32
  const float* targets     = (const float*)d_in[1];      // (64,1024) f32
  const float* lls         = (const float*)d_in[2];      // scalar f32
  const long long* pins    = (const long long*)d_in[3];  // (64,1024,2) i64
  float* out = (float*)d_out;

  char* ws = (char*)d_ws;
  float* Kws    = (float*)ws;                                    // 256 MB
  float* diffws = (float*)(ws + (size_t)BATCH * NN * NN * 4);    // 256 KB
  float* resws  = diffws + (size_t)BATCH * NN;                   // 256 B

  nppk_build_K<<<dim3(NBLK, NBLK, BATCH), dim3(16, 16), 0, stream>>>(
      pins, lls, Kws);
  nppk_build_diff<<<dim3(NN / 256, BATCH), 256, 0, stream>>>(
      preds, targets, pins, diffws);
  nppk_chol_solve<<<dim3(BATCH), 256, 0, stream>>>(Kws, diffws, resws);
  nppk_finalize<<<dim3(1), dim3(BATCH), 0, stream>>>(resws, out);
}